// GNN_57818849738867
// MI455X (gfx1250) — compile-verified
//
#include <hip/hip_runtime.h>
#include <hip/hip_bf16.h>

// ---------------------------------------------------------------------------
// CDNA5 (gfx1250) GCN forward: 4x (X@W -> sym-norm scatter-add -> ReLU),
// mean-pool per graph, 2-layer MLP head.
//   - GEMM: V_WMMA_F32_16X16X4_F32 (full fp32 precision on the matrix pipe),
//     A tile + fragment-order W staged in LDS (conflict-free wave32 access),
//     bias + self-loop epilogue fused into the GEMM store.
//   - Scatter: HW global_atomic_add_f32 (L2-resident: 51MB << 192MB L2).
// ---------------------------------------------------------------------------

typedef __attribute__((ext_vector_type(2))) float v2f;
typedef __attribute__((ext_vector_type(8))) float v8f;

#define NFEAT 128
#define PITCH_A  132   // floats: +4 pad -> strip-row float2 reads hit distinct banks
#define PITCH_B2 144   // float2s: 144 % 32 == 16 -> kg vs kg+1 rows 32 banks apart

static inline int cdiv_i(long long a, long long b) { return (int)((a + b - 1) / b); }

// --------------------------- utility kernels -------------------------------

__global__ __launch_bounds__(256) void zero_u32_kernel(unsigned* __restrict__ p, long long n) {
    long long i = (long long)blockIdx.x * 256 + threadIdx.x;
    if (i < n) p[i] = 0u;
}

__global__ __launch_bounds__(256) void deg_count_kernel(const int* __restrict__ dst,
                                                        int n_edges, int* __restrict__ deg) {
    int e = blockIdx.x * 256 + threadIdx.x;
    if (e < n_edges) atomicAdd(&deg[dst[e]], 1);
}

__global__ __launch_bounds__(256) void make_dinv_kernel(const int* __restrict__ deg,
                                                        float* __restrict__ dinv, int n_nodes) {
    int v = blockIdx.x * 256 + threadIdx.x;
    if (v < n_nodes) {
        // +1 self-loop guarantees deg >= 1
        dinv[v] = rsqrtf((float)deg[v] + 1.0f);
    }
}

// --------------------------- WMMA fp32 GEMM --------------------------------
// Per 256-thread block: 8 waves x 16-row strips over a 128-row tile.
//   H[r,:]   = relu?(act[r,:]) @ W
//   out[r,:] = bias + H[r,:] * dinv[r]^2          (self-loop msg, fused epilogue)
// A-frag (16x4 f32): lane l: m=l&15, half=l>>4; a = A[m][k0+2*half .. +1].
// B-frag: lane reads float2 (W[ka][col], W[ka+1][col]) -> one ds_load_b64.
// C/D (16x16 f32): vgpr r -> M = r + 8*half, N = l&15.

__global__ __launch_bounds__(256) void gcn_gemm_wmma_kernel(
    const float* __restrict__ X, const float* __restrict__ W,
    const float* __restrict__ dinv, const float* __restrict__ bias,
    float* __restrict__ H, float* __restrict__ out,
    int n_rows, int relu_in)
{
    __shared__ float sA[128 * PITCH_A];        // 67.5 KB
    __shared__ float sB[64 * PITCH_B2 * 2];    // 72 KB, W in fragment order (float2)

    const int t     = threadIdx.x;
    const int row0g = blockIdx.x * 128;

    // Stage W (128x128) -> LDS pre-swizzled: sB[kg*PITCH_B2 + col] = {W[2kg][col], W[2kg+1][col]}
    #pragma unroll
    for (int i = 0; i < 16; ++i) {
        int idx4 = t + i * 256;                // float4 index, 4096 total
        int r    = idx4 >> 5;
        int c    = (idx4 & 31) << 2;
        const float4 w4 = *(const float4*)(W + r * NFEAT + c);
        float* base = sB + (size_t)(r >> 1) * (PITCH_B2 * 2) + (r & 1);
        base[(c + 0) * 2] = w4.x;
        base[(c + 1) * 2] = w4.y;
        base[(c + 2) * 2] = w4.z;
        base[(c + 3) * 2] = w4.w;
    }
    // Stage A tile (128 rows) -> LDS, fusing ReLU of the previous layer
    #pragma unroll
    for (int i = 0; i < 16; ++i) {
        int idx4 = t + i * 256;
        int r    = idx4 >> 5;
        int c    = (idx4 & 31) << 2;
        int gr   = row0g + r;
        if (gr >= n_rows) gr = n_rows - 1;     // clamp; OOB rows never stored
        float4 a4 = *(const float4*)(X + (size_t)gr * NFEAT + c);
        if (relu_in) {
            a4.x = fmaxf(a4.x, 0.0f); a4.y = fmaxf(a4.y, 0.0f);
            a4.z = fmaxf(a4.z, 0.0f); a4.w = fmaxf(a4.w, 0.0f);
        }
        float* d = &sA[r * PITCH_A + c];
        d[0] = a4.x; d[1] = a4.y; d[2] = a4.z; d[3] = a4.w;
    }
    // Speculative prefetch of the next block's A tile (global_prefetch_b8)
    {
        int pr = row0g + 128 + (t >> 1);
        if (pr < n_rows)
            __builtin_prefetch(X + (size_t)pr * NFEAT + (t & 1) * 64, 0, 0);
    }
    __syncthreads();

    const int wave = t >> 5;
    const int lane = t & 31;
    const int half = lane >> 4;
    const int m16  = lane & 15;
    const int row0 = row0g + wave * 16;
    if (row0 >= n_rows) return;                // wave-uniform: EXEC all-1 inside

    v8f acc[8] = {};                           // 8 N-tiles of 16x16 f32

    for (int k0 = 0; k0 < NFEAT; k0 += 4) {
        const int ka = k0 + 2 * half;
        const int kg = ka >> 1;
        v2f a = *(const v2f*)(&sA[(wave * 16 + m16) * PITCH_A + ka]);
        const float* brow = sB + (size_t)kg * (PITCH_B2 * 2);
        #pragma unroll
        for (int nt = 0; nt < 8; ++nt) {
            v2f b = *(const v2f*)(brow + (nt * 16 + m16) * 2);
            acc[nt] = __builtin_amdgcn_wmma_f32_16x16x4_f32(
                /*neg_a=*/false, a, /*neg_b=*/false, b,
                /*c_mod=*/(short)0, acc[nt], /*reuse_a=*/false, /*reuse_b=*/false);
        }
    }

    // Fused epilogue: H = result; out = bias + result * dinv^2
    if (row0 + 16 <= n_rows) {                 // wave-uniform fast path (always, 100000%16==0)
        float dsq[8];
        #pragma unroll
        for (int r = 0; r < 8; ++r) {
            float di = dinv[row0 + r + 8 * half];
            dsq[r] = di * di;
        }
        #pragma unroll
        for (int nt = 0; nt < 8; ++nt) {
            const int col = nt * 16 + m16;
            const float bc = bias[col];
            #pragma unroll
            for (int r = 0; r < 8; ++r) {
                const size_t idx = (size_t)(row0 + r + 8 * half) * NFEAT + col;
                const float h = acc[nt][r];
                H[idx]   = h;
                out[idx] = bc + h * dsq[r];
            }
        }
    } else {                                   // guarded tail (unused at these sizes)
        #pragma unroll
        for (int nt = 0; nt < 8; ++nt) {
            const int col = nt * 16 + m16;
            const float bc = bias[col];
            #pragma unroll
            for (int r = 0; r < 8; ++r) {
                int row = row0 + r + 8 * half;
                if (row < n_rows) {
                    const size_t idx = (size_t)row * NFEAT + col;
                    const float h = acc[nt][r];
                    float di = dinv[row];
                    H[idx]   = h;
                    out[idx] = bc + h * di * di;
                }
            }
        }
    }
}

// ----------------------------- edge scatter --------------------------------
// out[dst,:] += H[src,:] * (dinv[src]*dinv[dst])  -- one wave per edge,
// lane q handles a float4 feature quad; HW f32 atomics stay in L2.

__global__ __launch_bounds__(256) void edge_scatter_kernel(
    const int* __restrict__ src, const int* __restrict__ dst,
    const float* __restrict__ dinv, const float* __restrict__ H,
    float* __restrict__ out, int n_edges)
{
    long long tid = (long long)blockIdx.x * 256 + threadIdx.x;
    int e = (int)(tid >> 5);
    int q = (int)(tid & 31);
    if (e >= n_edges) return;
    int s = src[e];
    int d = dst[e];
    float nrm = dinv[s] * dinv[d];
    float4 h4 = *(const float4*)(H + (size_t)s * NFEAT + q * 4);
    float* o = out + (size_t)d * NFEAT + q * 4;
    unsafeAtomicAdd(o + 0, h4.x * nrm);
    unsafeAtomicAdd(o + 1, h4.y * nrm);
    unsafeAtomicAdd(o + 2, h4.z * nrm);
    unsafeAtomicAdd(o + 3, h4.w * nrm);
}

// ----------------------------- pooling + MLP -------------------------------

__global__ __launch_bounds__(256) void pool_kernel(
    const float* __restrict__ act, const int* __restrict__ batch,
    float* __restrict__ pooled, int* __restrict__ cnt, int n_nodes)
{
    long long tid = (long long)blockIdx.x * 256 + threadIdx.x;
    int v = (int)(tid >> 5);
    int q = (int)(tid & 31);
    if (v >= n_nodes) return;
    int g = batch[v];
    float4 a = *(const float4*)(act + (size_t)v * NFEAT + q * 4);
    a.x = fmaxf(a.x, 0.0f); a.y = fmaxf(a.y, 0.0f);    // ReLU of last conv
    a.z = fmaxf(a.z, 0.0f); a.w = fmaxf(a.w, 0.0f);
    float* p = pooled + (size_t)g * NFEAT + q * 4;
    unsafeAtomicAdd(p + 0, a.x);
    unsafeAtomicAdd(p + 1, a.y);
    unsafeAtomicAdd(p + 2, a.z);
    unsafeAtomicAdd(p + 3, a.w);
    if (q == 0) atomicAdd(&cnt[g], 1);
}

__global__ __launch_bounds__(128) void mlp_kernel(
    const float* __restrict__ pooled, const int* __restrict__ cnt,
    const float* __restrict__ Ws1, const float* __restrict__ bs1,
    const float* __restrict__ Ws2, const float* __restrict__ bs2,
    float* __restrict__ out)
{
    __shared__ float p[128];
    __shared__ float red[128];
    int g = blockIdx.x;
    int t = threadIdx.x;
    float c = fmaxf((float)cnt[g], 1.0f);
    p[t] = pooled[(size_t)g * NFEAT + t] / c;
    __syncthreads();
    float s = bs1[t];
    #pragma unroll 8
    for (int i = 0; i < NFEAT; ++i) s += p[i] * Ws1[i * NFEAT + t];
    red[t] = fmaxf(s, 0.0f) * Ws2[t];
    __syncthreads();
    for (int w = 64; w > 0; w >>= 1) {
        if (t < w) red[t] += red[t + w];
        __syncthreads();
    }
    if (t == 0) out[g] = red[0] + bs2[0];
}

// ------------------------------- launcher ----------------------------------

extern "C" void kernel_launch(void* const* d_in, const int* in_sizes, int n_in,
                              void* d_out, int out_size, void* d_ws, size_t ws_size,
                              hipStream_t stream) {
    const int NN = in_sizes[0] / NFEAT;     // 100000 nodes
    const int NE = in_sizes[1] / 2;         // 600000 edges
    const int NG = out_size;                // 256 graphs

    const float* x     = (const float*)d_in[0];
    const int*   src   = (const int*)d_in[1];
    const int*   dst   = src + NE;
    const int*   batch = (const int*)d_in[2];
    const float* Wl[4] = {(const float*)d_in[3], (const float*)d_in[5],
                          (const float*)d_in[7], (const float*)d_in[9]};
    const float* bl[4] = {(const float*)d_in[4], (const float*)d_in[6],
                          (const float*)d_in[8], (const float*)d_in[10]};
    const float* Ws1 = (const float*)d_in[11];
    const float* bs1 = (const float*)d_in[12];
    const float* Ws2 = (const float*)d_in[13];
    const float* bs2 = (const float*)d_in[14];

    // Workspace carve-out (256B aligned): 2 activation buffers + small scratch
    char*  ws  = (char*)d_ws;
    size_t off = 0;
    auto carve = [&](size_t bytes) -> void* {
        void* p = ws + off;
        off = (off + bytes + 255) & ~(size_t)255;
        return p;
    };
    float* buf0   = (float*)carve((size_t)NN * NFEAT * sizeof(float)); // H (GEMM out)
    float* buf1   = (float*)carve((size_t)NN * NFEAT * sizeof(float)); // conv out / next act
    int*   deg    = (int*)  carve((size_t)NN * sizeof(int));
    float* dinv   = (float*)carve((size_t)NN * sizeof(float));
    float* pooled = (float*)carve((size_t)NG * NFEAT * sizeof(float)); // + cnt adjacent
    int*   cnt    = (int*)  carve((size_t)NG * sizeof(int));
    (void)ws_size; (void)n_in;

    // Zero accumulators (deterministic across graph replays)
    zero_u32_kernel<<<cdiv_i(NN, 256), 256, 0, stream>>>((unsigned*)deg, NN);
    zero_u32_kernel<<<cdiv_i((long long)NG * NFEAT + NG, 256), 256, 0, stream>>>(
        (unsigned*)pooled, (long long)NG * NFEAT + NG);   // pooled + cnt contiguous

    // Degrees (+ self-loop) -> dinv
    deg_count_kernel<<<cdiv_i(NE, 256), 256, 0, stream>>>(dst, NE, deg);
    make_dinv_kernel<<<cdiv_i(NN, 256), 256, 0, stream>>>(deg, dinv, NN);

    const int gemm_blocks = cdiv_i(NN, 128);
    const int node_quads  = cdiv_i((long long)NN * 32, 256);
    const int edge_quads  = cdiv_i((long long)NE * 32, 256);

    const float* act = x;
    for (int l = 0; l < 4; ++l) {
        // buf0 = H = relu?(act) @ W ; buf1 = bias + H*dinv^2 (self-loop init)
        gcn_gemm_wmma_kernel<<<gemm_blocks, 256, 0, stream>>>(
            act, Wl[l], dinv, bl[l], buf0, buf1, NN, l > 0);
        // buf1 += scatter-add of normalized edge messages (gather from buf0)
        edge_scatter_kernel<<<edge_quads, 256, 0, stream>>>(src, dst, dinv, buf0, buf1, NE);
        act = buf1;
    }

    pool_kernel<<<node_quads, 256, 0, stream>>>(buf1, batch, pooled, cnt, NN);
    mlp_kernel<<<NG, 128, 0, stream>>>(pooled, cnt, Ws1, bs1, Ws2, bs2, (float*)d_out);
}